// Model_39462159516149
// MI455X (gfx1250) — compile-verified
//
#include <hip/hip_runtime.h>

// ---------------------------------------------------------------------------
// Affine-scan kernel for MI455X (gfx1250).
// B points scanned over T steps; each step gathers 1 of 16 affine transforms.
// Memory-bound: ~134 MB traffic -> ~5.8 us at 23.3 TB/s. Strategy:
//   * 1 point per lane, scan state (p0,p1) in VGPRs
//   * 16-entry transform table in LDS (2 x float4 per entry)
//   * code[] staged to LDS via gfx1250 async global->LDS loads, double-buffered
//   * merged 96-bit output stores, 16x unrolled for deep store pipelining
// ---------------------------------------------------------------------------

#define BT    64     // threads per block (points per block) -> 2 waves
#define CHUNK 16     // timesteps of code staged per LDS chunk

#if defined(__gfx1250__) && __has_builtin(__builtin_amdgcn_global_load_async_to_lds_b32)
#define HAVE_ASYNC_LDS 1
#else
#define HAVE_ASYNC_LDS 0
#endif

#if HAVE_ASYNC_LDS
typedef __attribute__((address_space(1))) int g_int;   // global
typedef __attribute__((address_space(3))) int l_int;   // LDS
#endif

struct F3 { float x, y, z; };   // 12-byte packed output element

__device__ __forceinline__ void wait_async_zero() {
#if HAVE_ASYNC_LDS
#if __has_builtin(__builtin_amdgcn_s_wait_asynccnt)
    __builtin_amdgcn_s_wait_asynccnt(0);
#else
    asm volatile("s_wait_asynccnt 0" ::: "memory");
#endif
#endif
}

// Stage CHUNK rows of code (BT ints each) into the LDS tile `dst`.
__device__ __forceinline__ void stage_chunk(const int* __restrict__ code,
                                            int* dst, int t0, int B,
                                            int gb0, int tid) {
#if HAVE_ASYNC_LDS
#pragma unroll
    for (int i = 0; i < CHUNK; ++i) {
        const int* g = code + (t0 + i) * B + gb0 + tid;
        __builtin_amdgcn_global_load_async_to_lds_b32(
            (g_int*)(g),
            (l_int*)(dst + i * BT + tid),
            0, 0);
    }
#else
#pragma unroll
    for (int i = 0; i < CHUNK; ++i)
        dst[i * BT + tid] = code[(t0 + i) * B + gb0 + tid];
#endif
}

__global__ __launch_bounds__(BT) void affine_scan_kernel(
    const float* __restrict__ init_points,   // (B,2)
    const float* __restrict__ weights,       // (16,2,2)
    const float* __restrict__ biases,        // (16,2)
    const float* __restrict__ opacities,     // (16,)
    const int*   __restrict__ code,          // (T,B)
    float*       __restrict__ out,           // (T*B,3)
    int B, int T)
{
    __shared__ float4 tbl[32];               // entry n: [2n]={w00,w01,w10,w11} [2n+1]={b0,b1,op,0}
    __shared__ int    scode[2][CHUNK * BT];  // double-buffered code tile

    const int tid = threadIdx.x;
    const int gb0 = blockIdx.x * BT;
    const int b   = gb0 + tid;

    if (tid < 16) {
        const float* w = weights + tid * 4;
        tbl[2 * tid]     = make_float4(w[0], w[1], w[2], w[3]);
        tbl[2 * tid + 1] = make_float4(biases[2 * tid], biases[2 * tid + 1],
                                       opacities[tid], 0.0f);
    }

    float p0 = init_points[2 * b + 0];
    float p1 = init_points[2 * b + 1];

    // Prefetch chunk 0, then make it (and the table) visible block-wide.
    stage_chunk(code, &scode[0][0], 0, B, gb0, tid);
    wait_async_zero();
    __syncthreads();

    F3* __restrict__ out3 = (F3*)out;
    const int nchunk = T / CHUNK;

    for (int c = 0; c < nchunk; ++c) {
        // Kick off the next chunk's async copy while we consume this one.
        if (c + 1 < nchunk)
            stage_chunk(code, &scode[(c + 1) & 1][0], (c + 1) * CHUNK, B, gb0, tid);

        const int  t0 = c * CHUNK;
        const int* sc = &scode[c & 1][0];

#pragma unroll
        for (int i = 0; i < CHUNK; ++i) {
            const int idx = sc[i * BT + tid];
            const float4 wv = tbl[2 * idx];      // ds_load_b128
            const float4 bv = tbl[2 * idx + 1];  // ds_load_b128

            const float q0 = fmaf(wv.x, p0, fmaf(wv.y, p1, bv.x));
            const float q1 = fmaf(wv.z, p0, fmaf(wv.w, p1, bv.y));
            p0 = q0;
            p1 = q1;

            F3 o;
            o.x = fmaf(q0, 0.5f, 0.5f);
            o.y = fmaf(q1, 0.5f, 0.5f);
            o.z = bv.z;
            out3[(t0 + i) * B + b] = o;          // 96-bit merged store
        }

        // Ensure the prefetch landed in LDS, then publish block-wide before
        // anyone reads the other buffer (and before this buffer is reused).
        wait_async_zero();
        __syncthreads();
    }
}

extern "C" void kernel_launch(void* const* d_in, const int* in_sizes, int n_in,
                              void* d_out, int out_size, void* d_ws, size_t ws_size,
                              hipStream_t stream) {
    const float* init_points = (const float*)d_in[0];
    const float* weights     = (const float*)d_in[1];
    const float* biases      = (const float*)d_in[2];
    const float* opacities   = (const float*)d_in[3];
    const int*   code        = (const int*)d_in[4];

    const int B = in_sizes[0] / 2;        // 16384
    const int T = in_sizes[4] / B;        // 512

    dim3 grid(B / BT), block(BT);
    affine_scan_kernel<<<grid, block, 0, stream>>>(
        init_points, weights, biases, opacities, code, (float*)d_out, B, T);
}